// TensoRFEncoder_65652870087346
// MI455X (gfx1250) — compile-verified
//
#include <hip/hip_runtime.h>
#include <stdint.h>

#define CCH      48
#define RES      320
#define PLSZ     (RES * RES)      // 102400 floats per channel plane
#define LSTRIDE  322              // padded LDS row stride (banks: 322%64==2 -> conflict-free)
#define TPB      512
#define NBLK     2048

struct __attribute__((packed, aligned(4))) F2 { float x, y; };

// Per-point, per-plane bilinear setup with zero-padding folded into shifted weights.
// Returns two row offsets (pair loads at x=xa, xa+1) and 4 combined corner weights.
__device__ __forceinline__ void mkplane(float gx, float gy,
                                        int& oA, int& oB,
                                        float& q00, float& q01, float& q10, float& q11) {
    // ix = ((g+1)*320 - 1) * 0.5 = g*160 + 159.5
    float ix = fmaf(gx, 160.0f, 159.5f);
    float iy = fmaf(gy, 160.0f, 159.5f);
    float fx = floorf(ix), fy = floorf(iy);
    float wx = ix - fx,    wy = iy - fy;
    int x0 = (int)fx, y0 = (int)fy;

    // shifted pair weights: pair loaded at xa=clamp(x0,0,318) covers both edge cases
    float wA, wB;
    if (x0 < 0)         { wA = wx;          wB = 0.0f;        }  // only x=0 valid, gets wx
    else if (x0 >= 319) { wA = 0.0f;        wB = 1.0f - wx;   }  // only x=319 valid, gets 1-wx
    else                { wA = 1.0f - wx;   wB = wx;          }
    int xa = min(max(x0, 0), 318);

    float w0 = (y0 >= 0)  ? (1.0f - wy) : 0.0f;   // row y0 mask folded in
    float w1 = (y0 < 319) ? wy          : 0.0f;   // row y0+1 mask folded in
    int y0c = max(y0, 0);
    int y1c = min(y0 + 1, 319);

    oA = y0c * RES + xa;
    oB = y1c * RES + xa;
    q00 = wA * w0; q01 = wB * w0; q10 = wA * w1; q11 = wB * w1;
}

__device__ __forceinline__ void mkline(float t, int& o0, float& lA, float& lB) {
    float it = fmaf(t, 160.0f, 159.5f);
    float f0 = floorf(it);
    float wt = it - f0;
    int t0 = (int)f0;
    if (t0 < 0)         { lA = wt;          lB = 0.0f;        }
    else if (t0 >= 319) { lA = 0.0f;        lB = 1.0f - wt;   }
    else                { lA = 1.0f - wt;   lB = wt;          }
    o0 = min(max(t0, 0), 318);
}

__global__ __launch_bounds__(TPB)
void tensorf_encode_kernel(const float* __restrict__ coords,
                           const float* __restrict__ pyz,
                           const float* __restrict__ pzx,
                           const float* __restrict__ pxy,
                           const float* __restrict__ lx,
                           const float* __restrict__ ly,
                           const float* __restrict__ lz,
                           float* __restrict__ out, int n) {
    // All three line textures staged to LDS: 3*48*322*4 = 185,472 B (<320KB WGP LDS)
    __shared__ float smem[3 * CCH * LSTRIDE];

    // ---- Async global -> LDS staging (CDNA5 ASYNCcnt path) ----
    // 144 rows (line,channel) x 80 b128 segments per row
    for (int k = threadIdx.x; k < 3 * CCH * (RES / 4); k += TPB) {
        int row = k / (RES / 4);
        int seg = k - row * (RES / 4);
        int l   = row / CCH;
        int c   = row - l * CCH;
        const float* src = (l == 0 ? lx : (l == 1 ? ly : lz)) + (c * RES + seg * 4);
        uint32_t dst = (uint32_t)(uintptr_t)(&smem[row * LSTRIDE + seg * 4]);
        asm volatile("global_load_async_to_lds_b128 %0, %1, off"
                     :: "v"(dst), "v"(src) : "memory");
    }
    asm volatile("s_wait_asynccnt 0" ::: "memory");
    __syncthreads();

    const int slot = threadIdx.x >> 4;   // 0..31 : point slot within block
    const int lane = threadIdx.x & 15;   // 16 lanes cooperate on one point

    for (int p = blockIdx.x * (TPB / 16) + slot; p < n; p += gridDim.x * (TPB / 16)) {
        float x = coords[3 * p + 0];
        float y = coords[3 * p + 1];
        float z = coords[3 * p + 2];

        int   oA[3], oB[3], lo[3];
        float q00[3], q01[3], q10[3], q11[3], lA[3], lB[3];

        // f0: plane_yz(gx=y, gy=z) * line_x(x)
        // f1: plane_zx(gx=z, gy=x) * line_y(y)
        // f2: plane_xy(gx=x, gy=y) * line_z(z)
        mkplane(y, z, oA[0], oB[0], q00[0], q01[0], q10[0], q11[0]);
        mkplane(z, x, oA[1], oB[1], q00[1], q01[1], q10[1], q11[1]);
        mkplane(x, y, oA[2], oB[2], q00[2], q01[2], q10[2], q11[2]);
        mkline(x, lo[0], lA[0], lB[0]);
        mkline(y, lo[1], lA[1], lB[1]);
        mkline(z, lo[2], lA[2], lB[2]);

        #pragma unroll
        for (int k = 0; k < 3; ++k) {
            int c = lane + 16 * k;
            float acc = 0.0f;
            #pragma unroll
            for (int i = 0; i < 3; ++i) {
                const float* bp = (i == 0 ? pyz : (i == 1 ? pzx : pxy)) + (size_t)c * PLSZ;
                F2 a = *(const F2*)(bp + oA[i]);   // corners (y0, xa), (y0, xa+1)
                F2 b = *(const F2*)(bp + oB[i]);   // corners (y1, xa), (y1, xa+1)
                float r = a.x * q00[i] + a.y * q01[i] + b.x * q10[i] + b.y * q11[i];
                const float* sl = &smem[(i * CCH + c) * LSTRIDE + lo[i]];
                float s = sl[0] * lA[i] + sl[1] * lB[i];
                acc = fmaf(r, s, acc);
            }
            out[(size_t)p * CCH + c] = acc;   // 64B-contiguous per half-wave per k
        }
    }
}

extern "C" void kernel_launch(void* const* d_in, const int* in_sizes, int n_in,
                              void* d_out, int out_size, void* d_ws, size_t ws_size,
                              hipStream_t stream) {
    const float* coords = (const float*)d_in[0];
    const float* pyz    = (const float*)d_in[1];
    const float* pzx    = (const float*)d_in[2];
    const float* pxy    = (const float*)d_in[3];
    const float* lx     = (const float*)d_in[4];
    const float* ly     = (const float*)d_in[5];
    const float* lz     = (const float*)d_in[6];
    float* out = (float*)d_out;

    int n = in_sizes[0] / 3;   // number of points

    tensorf_encode_kernel<<<NBLK, TPB, 0, stream>>>(coords, pyz, pzx, pxy, lx, ly, lz, out, n);
}